// LearnableHighpass_78709570666738
// MI455X (gfx1250) — compile-verified
//
#include <hip/hip_runtime.h>
#include <math.h>
#include <stdint.h>

// ---------------------------------------------------------------------------
// Problem geometry
// ---------------------------------------------------------------------------
#define B_ROWS 64
#define T_LEN  524288
#define L_CHUNK 256                    // samples per chunk (per thread)
#define CPR (T_LEN / L_CHUNK)          // 2048 chunks per row
#define WG 256                         // threads per workgroup (8 wave32)
#define WGS_PER_ROW (CPR / WG)         // 8
#define G_GROUP 64                     // chunks per scan group
#define NG (CPR / G_GROUP)             // 32 groups per row
#define S_SLICE 32                     // samples per LDS slice
#define NSLICE (L_CHUNK / S_SLICE)     // 8
#define F4S (S_SLICE / 4)              // 8 float4 groups per slice

// ---------------------------------------------------------------------------
// CDNA5 async global->LDS path selection
//   1 = clang builtins, 2 = inline asm (gfx1250), 0 = plain loads (host pass)
// ---------------------------------------------------------------------------
#if defined(__HIP_DEVICE_COMPILE__) && \
    __has_builtin(__builtin_amdgcn_global_load_async_to_lds_b128) && \
    __has_builtin(__builtin_amdgcn_s_wait_asynccnt)
#define USE_ASYNC 1
#elif defined(__HIP_DEVICE_COMPILE__) && defined(__gfx1250__)
#define USE_ASYNC 2
#else
#define USE_ASYNC 0
#endif

#if USE_ASYNC == 1
typedef int v4i __attribute__((ext_vector_type(4)));
typedef __attribute__((address_space(1))) v4i GV4;   // global (device) v4i
typedef __attribute__((address_space(3))) v4i LV4;   // LDS (shared) v4i
__device__ __forceinline__ void async_load_b128(const float* gp, void* lp) {
  __builtin_amdgcn_global_load_async_to_lds_b128((GV4*)gp, (LV4*)lp, 0, 0);
}
template <int N> __device__ __forceinline__ void async_wait() {
  __builtin_amdgcn_s_wait_asynccnt(N);
}
#elif USE_ASYNC == 2
__device__ __forceinline__ void async_load_b128(const float* gp, void* lp) {
  unsigned laddr = (unsigned)(size_t)(__attribute__((address_space(3))) char*)lp;
  asm volatile("global_load_async_to_lds_b128 %0, %1, off"
               :: "v"(laddr), "v"(gp) : "memory");
}
template <int N> __device__ __forceinline__ void async_wait() {
  asm volatile("s_wait_asynccnt %0" :: "n"(N));
}
#endif

// Non-temporal 16B store via a native clang vector type (HIP float4 is a
// struct, which __builtin_nontemporal_store rejects). Keeps the 128 MB y
// stream from evicting x out of the 192 MB L2 between stage 1 and stage 3.
typedef float vf4 __attribute__((ext_vector_type(4)));
__device__ __forceinline__ void nt_store_f4(float* p, float a, float b,
                                            float c, float d) {
  vf4 v = {a, b, c, d};
#if __has_builtin(__builtin_nontemporal_store)
  __builtin_nontemporal_store(v, (vf4*)p);
#else
  *(vf4*)p = v;
#endif
}

// ---------------------------------------------------------------------------
// Biquad coefficients (torchaudio highpass_biquad, normalized by a0)
// ---------------------------------------------------------------------------
struct Coef { float b0, b1, b2, a1, a2; };

__device__ __forceinline__ Coef make_coefs(const float* ff, const float* fq,
                                           const int* sr) {
  float w0 = 6.28318530717958647692f * ff[0] / (float)sr[0];
  float s = sinf(w0), c = cosf(w0);
  float alpha = s / (2.0f * fq[0]);
  float a0 = 1.0f + alpha;
  Coef k;
  k.b0 = (0.5f * (1.0f + c)) / a0;
  k.b1 = (-(1.0f + c)) / a0;
  k.b2 = k.b0;
  k.a1 = (-2.0f * c) / a0;
  k.a2 = (1.0f - alpha) / a0;
  return k;
}

// 2x2 companion-matrix algebra for the linear state recurrence
struct M2 { float m00, m01, m10, m11; };
__device__ __forceinline__ M2 msq(M2 x) {
  M2 r;
  r.m00 = x.m00 * x.m00 + x.m01 * x.m10;
  r.m01 = x.m00 * x.m01 + x.m01 * x.m11;
  r.m10 = x.m10 * x.m00 + x.m11 * x.m10;
  r.m11 = x.m10 * x.m01 + x.m11 * x.m11;
  return r;
}
__device__ __forceinline__ M2 mpow2(M2 a, int k) {
  for (int i = 0; i < k; ++i) a = msq(a);
  return a;
}

// One DF-I biquad step; critical path = 1 FMA through y1.
#define BQSTEP(XIN, YOUT)                                            \
  do {                                                               \
    float f_ = fmaf(k.b0, (XIN), fmaf(k.b1, x1, k.b2 * x2));         \
    float y_ = fmaf(na1, y1, fmaf(na2, y2, f_));                     \
    x2 = x1; x1 = (XIN); y2 = y1; y1 = y_; (YOUT) = y_;              \
  } while (0)

// ---------------------------------------------------------------------------
// Stage 1 / Stage 3: per-chunk recurrence.
//   ZERO_INIT=true : zero-state pass, emit final state e = (y_{L-1}, y_{L-2})
//   ZERO_INIT=false: exact incoming state from cInit, write y (NT stores)
// Workgroup = 256 threads, each owns one 256-sample chunk; x is staged
// through LDS in 256-chunk x 32-sample double-buffered tiles via async copies.
// ---------------------------------------------------------------------------
template <bool ZERO_INIT>
__global__ __launch_bounds__(WG) void biquad_chunks(
    const float* __restrict__ x, const float* __restrict__ ff,
    const float* __restrict__ fq, const int* __restrict__ sr,
    const float2* __restrict__ cInit, float* __restrict__ out,
    float2* __restrict__ eOut) {
  const int tid = threadIdx.x;
  const int row = blockIdx.x / WGS_PER_ROW;
  const int cb = (blockIdx.x % WGS_PER_ROW) * WG;  // chunk base in row
  const int cg = cb + tid;                         // this thread's chunk
  const size_t rowBase = (size_t)row * T_LEN;
  const size_t t0 = (size_t)cg * L_CHUNK;

  Coef k = make_coefs(ff, fq, sr);
  const float na1 = -k.a1, na2 = -k.a2;

  float x1 = 0.f, x2 = 0.f;
  if (cg > 0) {
    x1 = x[rowBase + t0 - 1];
    x2 = x[rowBase + t0 - 2];
  }
  float y1 = 0.f, y2 = 0.f;
  if (!ZERO_INIT) {
    float2 s = cInit[row * CPR + cg];
    y1 = s.x;
    y2 = s.y;
  }

#if USE_ASYNC
  __shared__ float4 lds[2][F4S][WG];  // 64 KB double buffer
  const float* gtile = x + rowBase + (size_t)cb * L_CHUNK;

  // Issue one 256-chunk x 32-sample slice: 8 async b128 per lane.
  // Lanes 8i..8i+7 fetch one full 128B cacheline of chunk j; LDS dest
  // layout [ig][chunk] performs the transpose for conflict-free reads.
  auto issue = [&](int sl, int buf) {
#pragma unroll
    for (int kk = 0; kk < 8; ++kk) {
      int f = kk * WG + tid;
      int j = f >> 3;
      int ig = f & 7;
      const float* gp =
          gtile + (size_t)j * L_CHUNK + (size_t)sl * S_SLICE + ig * 4;
      async_load_b128(gp, &lds[buf][ig][j]);
    }
  };

  issue(0, 0);
  issue(1, 1);

  for (int sl = 0; sl < NSLICE; ++sl) {
    const int buf = sl & 1;
    if (sl < NSLICE - 1) async_wait<8>();  // oldest slice's 8 copies done
    else                 async_wait<0>();
    __syncthreads();  // all waves' copies for this slice are in LDS
#pragma unroll
    for (int ig = 0; ig < F4S; ++ig) {
      float4 xv = lds[buf][ig][tid];
      float4 yv;
      BQSTEP(xv.x, yv.x);
      BQSTEP(xv.y, yv.y);
      BQSTEP(xv.z, yv.z);
      BQSTEP(xv.w, yv.w);
      if (!ZERO_INIT) {
        nt_store_f4(out + rowBase + t0 + (size_t)sl * S_SLICE + ig * 4,
                    yv.x, yv.y, yv.z, yv.w);
      }
    }
    __syncthreads();  // everyone finished reading buf before refill
    if (sl + 2 < NSLICE) issue(sl + 2, buf);
  }
#else
  // Plain per-lane streaming float4 path (host pass / builtin-less fallback)
  for (int q4 = 0; q4 < L_CHUNK / 4; ++q4) {
    float4 xv =
        *reinterpret_cast<const float4*>(x + rowBase + t0 + (size_t)q4 * 4);
    float4 yv;
    BQSTEP(xv.x, yv.x);
    BQSTEP(xv.y, yv.y);
    BQSTEP(xv.z, yv.z);
    BQSTEP(xv.w, yv.w);
    if (!ZERO_INIT) {
      nt_store_f4(out + rowBase + t0 + (size_t)q4 * 4, yv.x, yv.y, yv.z, yv.w);
    }
  }
#endif

  if (ZERO_INIT) eOut[row * CPR + cg] = make_float2(y1, y2);
}

// ---------------------------------------------------------------------------
// Stage 2a: scan 64 chunk states per group: s <- M*s + e,  M = A^256
// ---------------------------------------------------------------------------
__global__ void scan_groups(const float2* __restrict__ e, const float* ff,
                            const float* fq, const int* sr,
                            float2* __restrict__ gFin) {
  int t = blockIdx.x * blockDim.x + threadIdx.x;  // 0..2047
  int r = t >> 5, g = t & 31;
  Coef k = make_coefs(ff, fq, sr);
  M2 A = {-k.a1, -k.a2, 1.f, 0.f};
  M2 M = mpow2(A, 8);  // A^256
  float s0 = 0.f, s1 = 0.f;
  int base = r * CPR + g * G_GROUP;
  for (int i = 0; i < G_GROUP; ++i) {
    float2 ev = e[base + i];
    float n0 = fmaf(M.m00, s0, fmaf(M.m01, s1, ev.x));
    float n1 = fmaf(M.m10, s0, fmaf(M.m11, s1, ev.y));
    s0 = n0; s1 = n1;
  }
  gFin[r * NG + g] = make_float2(s0, s1);
}

// Stage 2b: scan 32 group states per row: s <- Mg*s + gFin, Mg = A^16384
__global__ void scan_rows(const float2* __restrict__ gFin, const float* ff,
                          const float* fq, const int* sr,
                          float2* __restrict__ gInit) {
  int r = threadIdx.x;
  if (r >= B_ROWS) return;
  Coef k = make_coefs(ff, fq, sr);
  M2 A = {-k.a1, -k.a2, 1.f, 0.f};
  M2 Mg = mpow2(A, 14);  // A^(256*64)
  float s0 = 0.f, s1 = 0.f;
  for (int g = 0; g < NG; ++g) {
    gInit[r * NG + g] = make_float2(s0, s1);
    float2 ev = gFin[r * NG + g];
    float n0 = fmaf(Mg.m00, s0, fmaf(Mg.m01, s1, ev.x));
    float n1 = fmaf(Mg.m10, s0, fmaf(Mg.m11, s1, ev.y));
    s0 = n0; s1 = n1;
  }
}

// Stage 2c: broadcast exact incoming state to every chunk within its group
__global__ void bcast_init(const float2* __restrict__ e,
                           const float2* __restrict__ gInit, const float* ff,
                           const float* fq, const int* sr,
                           float2* __restrict__ cInit) {
  int t = blockIdx.x * blockDim.x + threadIdx.x;  // 0..2047
  int r = t >> 5, g = t & 31;
  Coef k = make_coefs(ff, fq, sr);
  M2 A = {-k.a1, -k.a2, 1.f, 0.f};
  M2 M = mpow2(A, 8);  // A^256
  float2 s = gInit[r * NG + g];
  float s0 = s.x, s1 = s.y;
  int base = r * CPR + g * G_GROUP;
  for (int i = 0; i < G_GROUP; ++i) {
    cInit[base + i] = make_float2(s0, s1);
    float2 ev = e[base + i];
    float n0 = fmaf(M.m00, s0, fmaf(M.m01, s1, ev.x));
    float n1 = fmaf(M.m10, s0, fmaf(M.m11, s1, ev.y));
    s0 = n0; s1 = n1;
  }
}

// Fallback if the harness workspace is too small: 64 fully sequential rows.
__global__ void biquad_naive(const float* __restrict__ x, const float* ff,
                             const float* fq, const int* sr,
                             float* __restrict__ y) {
  int r = blockIdx.x * blockDim.x + threadIdx.x;
  if (r >= B_ROWS) return;
  Coef k = make_coefs(ff, fq, sr);
  const float na1 = -k.a1, na2 = -k.a2;
  float x1 = 0.f, x2 = 0.f, y1 = 0.f, y2 = 0.f;
  size_t base = (size_t)r * T_LEN;
  for (int t = 0; t < T_LEN; ++t) {
    float xin = x[base + t];
    float yv;
    BQSTEP(xin, yv);
    y[base + t] = yv;
  }
}

// ---------------------------------------------------------------------------
// Launch
// ---------------------------------------------------------------------------
extern "C" void kernel_launch(void* const* d_in, const int* in_sizes, int n_in,
                              void* d_out, int out_size, void* d_ws,
                              size_t ws_size, hipStream_t stream) {
  const float* x = (const float*)d_in[0];
  const float* ff = (const float*)d_in[1];
  const float* fq = (const float*)d_in[2];
  const int* sr = (const int*)d_in[3];
  float* y = (float*)d_out;

  const size_t nE = (size_t)B_ROWS * CPR;      // chunk finals / inits
  const size_t nG = (size_t)B_ROWS * NG;       // group finals / inits
  const size_t need = (2 * nE + 2 * nG) * sizeof(float2);  // ~2.03 MB

  if (ws_size < need) {  // degenerate but correct path
    biquad_naive<<<2, 32, 0, stream>>>(x, ff, fq, sr, y);
    return;
  }

  float2* e = (float2*)d_ws;
  float2* cI = e + nE;
  float2* gF = cI + nE;
  float2* gI = gF + nG;

  // Stage 1: zero-state chunk finals (reads x once, 128 MB)
  biquad_chunks<true><<<B_ROWS * WGS_PER_ROW, WG, 0, stream>>>(
      x, ff, fq, sr, nullptr, nullptr, e);
  // Stage 2: hierarchical exact scan of 2x2 affine chunk maps
  scan_groups<<<(B_ROWS * NG) / WG, WG, 0, stream>>>(e, ff, fq, sr, gF);
  scan_rows<<<1, B_ROWS, 0, stream>>>(gF, ff, fq, sr, gI);
  bcast_init<<<(B_ROWS * NG) / WG, WG, 0, stream>>>(e, gI, ff, fq, sr, cI);
  // Stage 3: exact outputs (re-reads x — L2-resident thanks to NT y stores)
  biquad_chunks<false><<<B_ROWS * WGS_PER_ROW, WG, 0, stream>>>(
      x, ff, fq, sr, cI, y, nullptr);
}